// PositionalSparseLinear2d_79121887527367
// MI455X (gfx1250) — compile-verified
//
#include <hip/hip_runtime.h>
#include <cstdint>

// PositionalSparseLinear2d: out[b,o] = sum_k in[b, conn[o,k]] * w[o,k]
// B=64, O=512*512, K=8. Memory-bound random-gather; working set (144 MiB)
// is resident in MI455X's 192 MB L2. CDNA5 path used: async global->LDS
// gathers (ASYNCcnt) in a 4-stage software pipeline.

#define BATCH  64
#define N_IN   (512 * 512)
#define N_OUT  (512 * 512)
#define KC     8
#define TPB    256
#define STAGES 4   // pipeline depth (power of two)

#if defined(__HIP_DEVICE_COMPILE__) && defined(__gfx1250__)
  #define ON_GFX1250 1
#else
  #define ON_GFX1250 0
#endif

#if ON_GFX1250
  #define USE_ASYNC 1
  #if __has_builtin(__builtin_amdgcn_global_load_async_to_lds_b32)
    #define ASYNC_BUILTIN 1
  #else
    #define ASYNC_BUILTIN 0
  #endif
#else
  #define USE_ASYNC 0
  #define ASYNC_BUILTIN 0
#endif

// Builtin signature (from hipcc diagnostic): arg0 is addrspace(1) int*
// (non-const), arg1 is addrspace(3) int*, then imm offset + cpol.
typedef __attribute__((address_space(1))) int* gas1_ip;
typedef __attribute__((address_space(3))) int* gas3_ip;

// Issue one per-lane async gather: copies 4 bytes from global address gsrc to LDS address ldst.
// Tracked by ASYNCcnt, zero VGPR destinations.
__device__ __forceinline__ void async_gather_b32(const float* gsrc, float* ldst) {
#if USE_ASYNC
  #if ASYNC_BUILTIN
    __builtin_amdgcn_global_load_async_to_lds_b32(
        (gas1_ip)(uintptr_t)gsrc,
        (gas3_ip)(uint32_t)(uintptr_t)ldst,   // low 32 bits of flat LDS ptr are the LDS offset
        /*offset=*/0, /*cpol=*/0);
  #else
    unsigned loff = (unsigned)(uintptr_t)ldst;
    asm volatile("global_load_async_to_lds_b32 %0, %1, off"
                 :: "v"(loff), "v"(gsrc) : "memory");
  #endif
#else
  (void)gsrc; (void)ldst;
#endif
}

template <int N>
__device__ __forceinline__ void wait_asynccnt() {
#if USE_ASYNC
  #if __has_builtin(__builtin_amdgcn_s_wait_asynccnt)
    __builtin_amdgcn_s_wait_asynccnt(N);
  #else
    asm volatile("s_wait_asynccnt %0" :: "i"(N) : "memory");
  #endif
#endif
}

__global__ __launch_bounds__(TPB)
void PositionalSparseLinear2d_kernel(const float* __restrict__ in,
                                     const int*   __restrict__ conn,
                                     const float* __restrict__ wgt,
                                     float*       __restrict__ out) {
  const int t = (int)threadIdx.x;
  const int o = (int)blockIdx.x * TPB + t;

  // Coalesced b128 loads: this output's 8 indices and 8 weights (32 B each,
  // lane-consecutive -> the wave streams 1 KiB contiguously). Held in regs
  // for the whole 64-batch loop.
  const int4   c0 = ((const int4*)conn)[2 * o + 0];
  const int4   c1 = ((const int4*)conn)[2 * o + 1];
  const float4 w0 = ((const float4*)wgt)[2 * o + 0];
  const float4 w1 = ((const float4*)wgt)[2 * o + 1];
  int idx[KC] = {c0.x, c0.y, c0.z, c0.w, c1.x, c1.y, c1.z, c1.w};

#if USE_ASYNC
  // [stage][k][tid] layout: async writes and ds_load_b32 reads are both
  // stride-4B across lanes -> bank-conflict free. 32 KiB per block.
  __shared__ float buf[STAGES][KC][TPB];

  // Prologue: fill pipeline with STAGES-1 batches (24 gathers in flight).
  #pragma unroll
  for (int p = 0; p < STAGES - 1; ++p) {
    const float* gb = in + (size_t)p * N_IN;
    #pragma unroll
    for (int k = 0; k < KC; ++k)
      async_gather_b32(gb + idx[k], &buf[p][k][t]);
  }

  for (int b = 0; b < BATCH; ++b) {
    const int st = b & (STAGES - 1);
    const int nb = b + (STAGES - 1);
    if (nb < BATCH) {
      // Keep the pipeline full, then wait until only the 3 newer batches
      // (3*KC loads) remain outstanding => batch b's gathers have landed
      // (async loads retire in order per ISA ch.10.7).
      const float* gb = in + (size_t)nb * N_IN;
      const int ns = nb & (STAGES - 1);
      #pragma unroll
      for (int k = 0; k < KC; ++k)
        async_gather_b32(gb + idx[k], &buf[ns][k][t]);
      wait_asynccnt<KC * (STAGES - 1)>();
    } else {
      const int rem = (BATCH - 1) - b;   // batches still in flight beyond b
      if (rem >= 2)      wait_asynccnt<2 * KC>();
      else if (rem == 1) wait_asynccnt<KC>();
      else               wait_asynccnt<0>();
    }
    asm volatile("" ::: "memory");  // order LDS reads after the wait

    float s = buf[st][0][t] * w0.x;
    s = fmaf(buf[st][1][t], w0.y, s);
    s = fmaf(buf[st][2][t], w0.z, s);
    s = fmaf(buf[st][3][t], w0.w, s);
    s = fmaf(buf[st][4][t], w1.x, s);
    s = fmaf(buf[st][5][t], w1.y, s);
    s = fmaf(buf[st][6][t], w1.z, s);
    s = fmaf(buf[st][7][t], w1.w, s);
    out[(size_t)b * N_OUT + o] = s;   // coalesced b32 store
  }
#else
  // Fallback (host pass / non-gfx1250 targets): direct register gather; the
  // compiler pipelines the scattered loads via LOADcnt.
  #pragma unroll 2
  for (int b = 0; b < BATCH; ++b) {
    const float* gb = in + (size_t)b * N_IN;
    float s = gb[idx[0]] * w0.x;
    s = fmaf(gb[idx[1]], w0.y, s);
    s = fmaf(gb[idx[2]], w0.z, s);
    s = fmaf(gb[idx[3]], w0.w, s);
    s = fmaf(gb[idx[4]], w1.x, s);
    s = fmaf(gb[idx[5]], w1.y, s);
    s = fmaf(gb[idx[6]], w1.z, s);
    s = fmaf(gb[idx[7]], w1.w, s);
    out[(size_t)b * N_OUT + o] = s;
  }
#endif
}

extern "C" void kernel_launch(void* const* d_in, const int* in_sizes, int n_in,
                              void* d_out, int out_size, void* d_ws, size_t ws_size,
                              hipStream_t stream) {
  (void)in_sizes; (void)n_in; (void)out_size; (void)d_ws; (void)ws_size;
  const float* in   = (const float*)d_in[0];
  const int*   conn = (const int*)d_in[1];
  const float* wgt  = (const float*)d_in[2];
  float*       out  = (float*)d_out;

  dim3 grid(N_OUT / TPB);   // 1024 blocks x 256 threads = one thread per output
  dim3 block(TPB);
  PositionalSparseLinear2d_kernel<<<grid, block, 0, stream>>>(in, conn, wgt, out);
}